// DropGraph_24713241822120
// MI455X (gfx1250) — compile-verified
//
#include <hip/hip_runtime.h>
#include <cstdint>

typedef float v4f __attribute__((ext_vector_type(4)));

#define TPB 256
#define BX 64
#define NBATCH 64
#define NNODE 48
#define VEC_PER_BATCH (128 * 256 * 48 / 4)   /* 393216 float4 per batch */
#define STRIDE (TPB * BX)                    /* 16384 ; 16384 % 12 == 4 */
#define ITERS (VEC_PER_BATCH / STRIDE)       /* 24 (divisible by 3)     */

#if defined(__has_builtin)
#  if __has_builtin(__builtin_amdgcn_global_load_async_to_lds_b128)
#    define HAVE_ASYNC_LDS 1
#  endif
#endif

#ifdef HAVE_ASYNC_LDS
// Builtin signature (probe-confirmed): (<4 x i32> addrspace(1)*, <4 x i32> addrspace(3)*, imm, imm)
typedef int vi4 __attribute__((vector_size(16)));
typedef __attribute__((address_space(1))) vi4 gvi4;   // global (AS1)
typedef __attribute__((address_space(3))) vi4 lvi4;   // LDS    (AS3)

__device__ __forceinline__ void async_b128(const void* g, void* l) {
  __builtin_amdgcn_global_load_async_to_lds_b128(
      (gvi4*)(unsigned long long)(uintptr_t)g,
      (lvi4*)(unsigned int)(uintptr_t)l,
      0, 0);
}
#endif

// ---- Adjacency folded to a compile-time table: 48 x 64-bit rows ----
struct AdjTable { unsigned long long row[NNODE]; };

static constexpr AdjTable build_adj() {
  AdjTable t{};
  for (int i = 0; i < NNODE; ++i) t.row[i] = 1ull << i;            // self loops
  const int ha[23] = {0,0,0,0,0,1,2,3,5,6,7,9,10,11,13,14,15,17,18,19,5,9,13};
  const int hb[23] = {1,5,9,13,17,2,3,4,6,7,8,10,11,12,14,15,16,18,19,20,9,13,17};
  for (int off = 0; off <= 21; off += 21)
    for (int e = 0; e < 23; ++e) {
      int a = ha[e] + off, c = hb[e] + off;
      t.row[a] |= 1ull << c;
      t.row[c] |= 1ull << a;
    }
  const int pa[7] = {42,42,43,44,45,46,47};
  const int pb[7] = {43,44,45,46,47,0,21};
  for (int e = 0; e < 7; ++e) {
    t.row[pa[e]] |= 1ull << pb[e];
    t.row[pb[e]] |= 1ull << pa[e];
  }
  return t;
}

__constant__ AdjTable ADJT = build_adj();

__global__ void __launch_bounds__(TPB) dropgraph_kernel(
    const v4f* __restrict__ x, const float* __restrict__ drop_rand,
    const int* __restrict__ seed_idx, v4f* __restrict__ out) {
  __shared__ __align__(16) float s_dr[NBATCH];
  __shared__ __align__(16) int   s_si[NBATCH];
  __shared__ float s_inv;
  __shared__ __align__(16) float s_sc[NNODE];

  const int tid = threadIdx.x;
  const int b   = blockIdx.y;

  // ---- Stage metadata into LDS via CDNA5 async-to-LDS (cooperative fill) ----
#ifdef HAVE_ASYNC_LDS
  if (tid < 16)      async_b128(drop_rand + tid * 4,        s_dr + tid * 4);
  else if (tid < 32) async_b128(seed_idx + (tid - 16) * 4,  s_si + (tid - 16) * 4);
#  if __has_builtin(__builtin_amdgcn_s_wait_asynccnt)
  __builtin_amdgcn_s_wait_asynccnt(0);
#  else
  asm volatile("s_wait_asynccnt 0" ::: "memory");
#  endif
#else
  if (tid < NBATCH) { s_dr[tid] = drop_rand[tid]; s_si[tid] = seed_idx[tid]; }
#endif
  __syncthreads();

  // ---- Global keep ratio: wave-0 shuffle reduction over 64 batches ----
  if (tid < 32) {
    bool d0 = s_dr[tid] < 0.1f;
    bool d1 = s_dr[tid + 32] < 0.1f;
    int t = (NNODE - (d0 ? __popcll(ADJT.row[s_si[tid]]) : 0)) +
            (NNODE - (d1 ? __popcll(ADJT.row[s_si[tid + 32]]) : 0));
    for (int off = 16; off > 0; off >>= 1) t += __shfl_xor(t, off, 32);
    if (tid == 0) {
      float kr = (float)t / 3072.0f;          // mean over 64*48, like jnp.mean
      s_inv = 1.0f / ((t > 0) ? kr : 1.0f);
    }
  }
  __syncthreads();

  // ---- This batch's 48-entry scale row ----
  if (tid < NNODE) {
    bool drop = s_dr[b] < 0.1f;
    unsigned long long row = ADJT.row[s_si[b]];
    float m = (drop && ((row >> tid) & 1ull)) ? 0.0f : 1.0f;
    s_sc[tid] = m * s_inv;
  }
  __syncthreads();

  // ---- Streaming body: NT B128 load -> scale -> NT B128 store ----
  const v4f* s4v = (const v4f*)s_sc;           // 12 float4 (period of mask in vec4 units)
  const v4f* __restrict__ xb = x   + (size_t)b * VEC_PER_BATCH;
  v4f*       __restrict__ ob = out + (size_t)b * VEC_PER_BATCH;

  int p = blockIdx.x * TPB + tid;
  int j = p % 12;
  // STRIDE % 12 == 4 -> each thread cycles through only 3 scale vectors.
  v4f sa = s4v[j];
  v4f sb = s4v[(j + 4) % 12];
  v4f sc = s4v[(j + 8) % 12];

#pragma unroll 2
  for (int i = 0; i < ITERS; i += 3) {
    v4f va = __builtin_nontemporal_load(xb + p);
    v4f vb = __builtin_nontemporal_load(xb + p + STRIDE);
    v4f vc = __builtin_nontemporal_load(xb + p + 2 * STRIDE);
    __builtin_nontemporal_store(va * sa, ob + p);
    __builtin_nontemporal_store(vb * sb, ob + p + STRIDE);
    __builtin_nontemporal_store(vc * sc, ob + p + 2 * STRIDE);
    p += 3 * STRIDE;
  }
}

extern "C" void kernel_launch(void* const* d_in, const int* in_sizes, int n_in,
                              void* d_out, int out_size, void* d_ws, size_t ws_size,
                              hipStream_t stream) {
  (void)in_sizes; (void)n_in; (void)out_size; (void)d_ws; (void)ws_size;
  const v4f*   x  = (const v4f*)d_in[0];
  const float* dr = (const float*)d_in[1];
  const int*   si = (const int*)d_in[2];
  dim3 grid(BX, NBATCH);
  dropgraph_kernel<<<grid, TPB, 0, stream>>>(x, dr, si, (v4f*)d_out);
}